// Dummy_38774964748552
// MI455X (gfx1250) — compile-verified
//
#include <hip/hip_runtime.h>
#include <cstdint>

// mLSTM recurrence: T=2048 sequential steps, B=4096 independent lanes, H=5.
// One lane = one batch element; weights + state in registers (VGPR-forced).
// x-stream staged via CDNA5 async global->LDS loads (ASYNCcnt), double buffered.
// Gate GEMM packed into float2 pairs -> v_pk_fma_f32; activations via v_tanh_f32
// (in-place, hazard-safe blocks); sigmoid 0.5-prescale folded into i/f/o rows.

#define TLEN  2048
#define BATCH 4096
#define HID   5
#define TS    32              // timesteps per LDS tile
#define NTILES (TLEN / TS)    // 64

typedef float v2f __attribute__((ext_vector_type(2)));

#if __has_builtin(__builtin_amdgcn_tanhf)
  #define TANH1(x) __builtin_amdgcn_tanhf(x)
  #define HAVE_TANH_BUILTIN 1
#elif __has_builtin(__builtin_amdgcn_tanh_f32)
  #define TANH1(x) __builtin_amdgcn_tanh_f32(x)
  #define HAVE_TANH_BUILTIN 1
#else
  #define HAVE_TANH_BUILTIN 0
#endif

// 4 independent v_tanh_f32 computed IN PLACE (inputs are dead afterwards, so
// tied "+v" operands avoid any register copies). Each subsequent tanh is the
// required independent op for the previous one (TRANS hazard, ISA 7.4.3);
// a single trailing v_nop protects the last result.
__device__ __forceinline__ void tanh4_ip(float& x0, float& x1, float& x2, float& x3) {
#if HAVE_TANH_BUILTIN
  x0 = TANH1(x0); x1 = TANH1(x1); x2 = TANH1(x2); x3 = TANH1(x3);
#else
  asm("v_tanh_f32 %0, %0\n\t"
      "v_tanh_f32 %1, %1\n\t"
      "v_tanh_f32 %2, %2\n\t"
      "v_tanh_f32 %3, %3\n\t"
      "v_nop"
      : "+v"(x0), "+v"(x1), "+v"(x2), "+v"(x3));
#endif
}

// 5 tanhs with separate outputs (inputs c[] must survive to the next step).
__device__ __forceinline__ void tanh5(const float* x, float* r) {
#if HAVE_TANH_BUILTIN
#pragma unroll
  for (int u = 0; u < HID; ++u) r[u] = TANH1(x[u]);
#else
  asm("v_tanh_f32 %0, %5\n\t"
      "v_tanh_f32 %1, %6\n\t"
      "v_tanh_f32 %2, %7\n\t"
      "v_tanh_f32 %3, %8\n\t"
      "v_tanh_f32 %4, %9\n\t"
      "v_nop"
      : "=&v"(r[0]), "=&v"(r[1]), "=&v"(r[2]), "=&v"(r[3]), "=&v"(r[4])
      : "v"(x[0]), "v"(x[1]), "v"(x[2]), "v"(x[3]), "v"(x[4]));
#endif
}

__global__ void __launch_bounds__(32)
mlstm_kernel(const float* __restrict__ xb,  const float* __restrict__ h0,
             const float* __restrict__ c0,  const float* __restrict__ Wmx,
             const float* __restrict__ Wmh, const float* __restrict__ Wih,
             const float* __restrict__ Whh, const float* __restrict__ b_ih,
             const float* __restrict__ b_hh, float* __restrict__ out)
{
  __shared__ float smem[2][TS][32];   // [buf][timestep][lane], bank-conflict-free

  const int lane = threadIdx.x;       // 0..31
  const int b    = blockIdx.x * 32 + lane;

  // Opaque VGPR zero: adding it forces weight values into VGPRs (we run
  // ~1 wave/SIMD, VGPRs are plentiful; avoids SGPR spill writelane/readlane).
  float vzero;
  asm volatile("v_mov_b32 %0, 0" : "=v"(vzero));
  const v2f vz2 = {vzero, vzero};

  // ---- pack tiny weights into VGPR-resident registers ----
  float wmxr[HID];
#pragma unroll
  for (int u = 0; u < HID; ++u) wmxr[u] = Wmx[u];

  // Wmh rows as pairs {row 2p, row 2p+1} over v, plus scalar row 4.
  v2f   wmh2[2 * HID];       // [p*HID + v], p = 0..1
  float wmh4[HID];
#pragma unroll
  for (int p = 0; p < 2; ++p)
#pragma unroll
    for (int v = 0; v < HID; ++v)
      wmh2[p * HID + v] = (v2f){Wmh[(2 * p) * HID + v], Wmh[(2 * p + 1) * HID + v]} + vz2;
#pragma unroll
  for (int v = 0; v < HID; ++v) wmh4[v] = Wmh[4 * HID + v] + vzero;

  // Gate pairs: j = 2p, 2p+1 for p = 0..9.  Wmx folded into Whh.
  // Sigmoid rows (i: j=0..4, f: 5..9, o: 15..19) pre-scaled by 0.5 so that
  // sigmoid(z) = fma(0.5, tanh(z_scaled), 0.5) needs no inner multiply.
  v2f wih2[10], bc2[10], whhp[10 * HID];
#pragma unroll
  for (int p = 0; p < 10; ++p) {
    const int j0 = 2 * p, j1 = 2 * p + 1;
    const float s0 = (j0 >= 2 * HID && j0 < 3 * HID) ? 1.0f : 0.5f;  // g rows: 1.0
    const float s1 = (j1 >= 2 * HID && j1 < 3 * HID) ? 1.0f : 0.5f;
    wih2[p] = (v2f){Wih[j0] * s0, Wih[j1] * s1} + vz2;
    bc2[p]  = (v2f){(b_ih[j0] + b_hh[j0]) * s0,
                    (b_ih[j1] + b_hh[j1]) * s1} + vz2;
#pragma unroll
    for (int v = 0; v < HID; ++v)
      whhp[p * HID + v] = (v2f){Whh[j0 * HID + v] * wmxr[v] * s0,
                                Whh[j1 * HID + v] * wmxr[v] * s1} + vz2;
  }

  // ---- state ----
  float h[HID], c[HID];
#pragma unroll
  for (int u = 0; u < HID; ++u) { h[u] = h0[b * HID + u]; c[u] = c0[b * HID + u]; }

  // LDS byte offsets of this lane's column (low 32 bits of flat shared addr).
  unsigned lds_col[2];
  lds_col[0] = (unsigned)(uintptr_t)(&smem[0][0][lane]);
  lds_col[1] = (unsigned)(uintptr_t)(&smem[1][0][lane]);

  // ---- issue tile 0 (async global -> LDS, tracked by ASYNCcnt) ----
  {
    unsigned goff = (unsigned)(b) * 4u;         // t = 0
    unsigned lo   = lds_col[0];
#pragma unroll
    for (int ts = 0; ts < TS; ++ts) {
      asm volatile("global_load_async_to_lds_b32 %0, %1, %2"
                   :: "v"(lo), "v"(goff), "s"(xb) : "memory");
      lo   += 32u * 4u;
      goff += BATCH * 4u;
    }
  }

#pragma unroll 1
  for (int k = 0; k < NTILES; ++k) {
    // prefetch next tile into the other buffer, then wait for current tile
    if (k + 1 < NTILES) {
      unsigned goff = (unsigned)((k + 1) * TS * BATCH + b) * 4u;
      unsigned lo   = lds_col[(k + 1) & 1];
#pragma unroll
      for (int ts = 0; ts < TS; ++ts) {
        asm volatile("global_load_async_to_lds_b32 %0, %1, %2"
                     :: "v"(lo), "v"(goff), "s"(xb) : "memory");
        lo   += 32u * 4u;
        goff += BATCH * 4u;
      }
      // 32 newer loads outstanding; in-order completion => current tile done
      asm volatile("s_wait_asynccnt 32" ::: "memory");
    } else {
      asm volatile("s_wait_asynccnt 0" ::: "memory");
    }

    const int buf = k & 1;   // direct indexing keeps addrspace(3) -> ds_load

#pragma unroll 2
    for (int ts = 0; ts < TS; ++ts) {
      float x = smem[buf][ts][lane];

      // h broadcast pairs (backend folds splats via op_sel)
      v2f hb[HID];
#pragma unroll
      for (int v = 0; v < HID; ++v) hb[v] = (v2f){h[v], h[v]};

      // m'_u = x * (Wmh h)_u   (Wmx folded into whhp); rows packed in pairs
      v2f d01 = wmh2[0] * hb[0];
      v2f d23 = wmh2[HID] * hb[0];
      float d4 = wmh4[0] * h[0];
#pragma unroll
      for (int v = 1; v < HID; ++v) {
        d01 = wmh2[v] * hb[v] + d01;
        d23 = wmh2[HID + v] * hb[v] + d23;
        d4  = fmaf(wmh4[v], h[v], d4);
      }
      const v2f xv = {x, x};
      v2f m01 = xv * d01;
      v2f m23 = xv * d23;
      float m4 = x * d4;

      v2f mb[HID];
      mb[0] = (v2f){m01.x, m01.x};
      mb[1] = (v2f){m01.y, m01.y};
      mb[2] = (v2f){m23.x, m23.x};
      mb[3] = (v2f){m23.y, m23.y};
      mb[4] = (v2f){m4, m4};

      // gates: 10 independent packed FMA chains (v_pk_fma_f32)
      v2f gp[10];
#pragma unroll
      for (int p = 0; p < 10; ++p) {
        v2f a = xv * wih2[p] + bc2[p];
#pragma unroll
        for (int v = 0; v < HID; ++v) a = whhp[p * HID + v] * mb[v] + a;
        gp[p] = a;
      }

      float gt[20];
#pragma unroll
      for (int p = 0; p < 10; ++p) { gt[2 * p] = gp[p].x; gt[2 * p + 1] = gp[p].y; }

      // activations: 20 gate tanhs in 5 hazard-safe in-place blocks of 4
#pragma unroll
      for (int u = 0; u < HID; ++u)
        tanh4_ip(gt[u], gt[HID + u], gt[2 * HID + u], gt[3 * HID + u]);

      float og[HID];
#pragma unroll
      for (int u = 0; u < HID; ++u) {
        float ig = fmaf(0.5f, gt[u], 0.5f);
        float fg = fmaf(0.5f, gt[HID + u], 0.5f);
        og[u]    = fmaf(0.5f, gt[3 * HID + u], 0.5f);
        c[u] = fmaf(fg, c[u], ig * gt[2 * HID + u]);
      }

      float tc[HID];
      tanh5(c, tc);
#pragma unroll
      for (int u = 0; u < HID; ++u) h[u] = og[u] * tc[u];
    }
  }

  // out = relu(h_T), shape [B, H]
#pragma unroll
  for (int u = 0; u < HID; ++u) out[b * HID + u] = fmaxf(h[u], 0.0f);
}

extern "C" void kernel_launch(void* const* d_in, const int* in_sizes, int n_in,
                              void* d_out, int out_size, void* d_ws, size_t ws_size,
                              hipStream_t stream) {
  (void)in_sizes; (void)n_in; (void)out_size; (void)d_ws; (void)ws_size;
  const float* xb  = (const float*)d_in[0];
  const float* h0  = (const float*)d_in[1];
  const float* c0  = (const float*)d_in[2];
  const float* Wmx = (const float*)d_in[3];
  const float* Wmh = (const float*)d_in[4];
  const float* Wih = (const float*)d_in[5];
  const float* Whh = (const float*)d_in[6];
  const float* bih = (const float*)d_in[7];
  const float* bhh = (const float*)d_in[8];
  float* out = (float*)d_out;

  dim3 grid(BATCH / 32);
  dim3 block(32);
  mlstm_kernel<<<grid, block, 0, stream>>>(xb, h0, c0, Wmx, Wmh, Wih, Whh,
                                           bih, bhh, out);
}